// DiffLoss_77214922048106
// MI455X (gfx1250) — compile-verified
//
#include <hip/hip_runtime.h>

typedef float v2f __attribute__((ext_vector_type(2)));
typedef float v4f __attribute__((ext_vector_type(4)));
typedef float v8f __attribute__((ext_vector_type(8)));

#define ELEMS_PER_SLICE   (64 * 64 * 64)   // 262144 per (b,s)
#define NUM_SLICES        (16 * 12)        // 192
#define CHUNKS_PER_SLICE  16
#define FLOATS_PER_CHUNK  (ELEMS_PER_SLICE / CHUNKS_PER_SLICE)  // 16384
#define BLOCK             256
#define ITERS             (FLOATS_PER_CHUNK / (BLOCK * 4))      // 16
#define NBLOCKS           (NUM_SLICES * CHUNKS_PER_SLICE)       // 3072

// Pass 1: stream both tensors once with non-temporal 128-bit loads and reduce
// each 16384-float chunk with fp32 WMMA (A = ones, B = data -> D rows = column sums).
__global__ __launch_bounds__(BLOCK)
void diffloss_partial(const float* __restrict__ common,
                      const float* __restrict__ specific,
                      float* __restrict__ ws_c,
                      float* __restrict__ ws_s) {
    const int tid = threadIdx.x;
    const int blk = blockIdx.x;
    // Chunks tile the whole [B,S,C,H,W] volume contiguously.
    const size_t base = (size_t)blk * FLOATS_PER_CHUNK + (size_t)tid * 4;

    v2f ones;
    ones.x = 1.0f;
    ones.y = 1.0f;

    v8f ac0 = {};
    v8f ac1 = {};
    v8f as0 = {};
    v8f as1 = {};

    for (int it = 0; it < ITERS; ++it) {
        const size_t off = base + (size_t)it * (BLOCK * 4);
        v4f c = __builtin_nontemporal_load((const v4f*)(common + off));
        v4f s = __builtin_nontemporal_load((const v4f*)(specific + off));

        v2f c0; c0.x = c.x; c0.y = c.y;
        v2f c1; c1.x = c.z; c1.y = c.w;
        v2f s0; s0.x = s.x; s0.y = s.y;
        v2f s1; s1.x = s.z; s1.y = s.w;

        // D = ones(16x4) x data(4x16) + C : exact fp32 column-sum accumulation.
        ac0 = __builtin_amdgcn_wmma_f32_16x16x4_f32(false, ones, false, c0,
                                                    (short)0, ac0, false, false);
        as0 = __builtin_amdgcn_wmma_f32_16x16x4_f32(false, ones, false, s0,
                                                    (short)0, as0, false, false);
        ac1 = __builtin_amdgcn_wmma_f32_16x16x4_f32(false, ones, false, c1,
                                                    (short)0, ac1, false, false);
        as1 = __builtin_amdgcn_wmma_f32_16x16x4_f32(false, ones, false, s1,
                                                    (short)0, as1, false, false);
    }

    // Per lane, D[0] holds one column-sum; columns are replicated across the
    // two 16-lane halves, so a 32-lane sum double-counts the wave total.
    float vc = ac0[0] + ac1[0];
    float vs = as0[0] + as1[0];
    #pragma unroll
    for (int o = 16; o > 0; o >>= 1) {
        vc += __shfl_xor(vc, o, 32);
        vs += __shfl_xor(vs, o, 32);
    }

    __shared__ float redc[BLOCK / 32];
    __shared__ float reds[BLOCK / 32];
    if ((tid & 31) == 0) {
        redc[tid >> 5] = vc;
        reds[tid >> 5] = vs;
    }
    __syncthreads();
    if (tid == 0) {
        float tc = 0.0f, ts = 0.0f;
        #pragma unroll
        for (int w = 0; w < BLOCK / 32; ++w) {
            tc += redc[w];
            ts += reds[w];
        }
        ws_c[blk] = tc * 0.5f;  // undo the 2x lane replication
        ws_s[blk] = ts * 0.5f;
    }
}

// Pass 2: fold 16 chunk-partials per slice, form Sc*Ss per slice, reduce 192
// slices, apply exact normalization 1/(N^2 * 192).
__global__ __launch_bounds__(256)
void diffloss_finalize(const float* __restrict__ ws_c,
                       const float* __restrict__ ws_s,
                       float* __restrict__ out) {
    const int t = threadIdx.x;
    float p = 0.0f;
    if (t < NUM_SLICES) {
        float sc = 0.0f, ss = 0.0f;
        #pragma unroll
        for (int j = 0; j < CHUNKS_PER_SLICE; ++j) {
            sc += ws_c[t * CHUNKS_PER_SLICE + j];
            ss += ws_s[t * CHUNKS_PER_SLICE + j];
        }
        p = sc * ss;
    }
    #pragma unroll
    for (int o = 16; o > 0; o >>= 1) {
        p += __shfl_xor(p, o, 32);
    }
    __shared__ float red[8];
    if ((t & 31) == 0) red[t >> 5] = p;
    __syncthreads();
    if (t == 0) {
        float total = 0.0f;
        #pragma unroll
        for (int w = 0; w < 8; ++w) total += red[w];
        // 262144^2 = 2^36 and *192 = 3*2^42: both exact in fp32.
        const float kScale =
            1.0f / (262144.0f * 262144.0f * (float)NUM_SLICES);
        out[0] = total * kScale;
    }
}

extern "C" void kernel_launch(void* const* d_in, const int* in_sizes, int n_in,
                              void* d_out, int out_size, void* d_ws, size_t ws_size,
                              hipStream_t stream) {
    const float* common   = (const float*)d_in[0];
    const float* specific = (const float*)d_in[1];
    float* ws_c = (float*)d_ws;            // NBLOCKS floats
    float* ws_s = ws_c + NBLOCKS;          // NBLOCKS floats (24.6 KB total)

    diffloss_partial<<<NBLOCKS, BLOCK, 0, stream>>>(common, specific, ws_c, ws_s);
    diffloss_finalize<<<1, 256, 0, stream>>>(ws_c, ws_s, (float*)d_out);
}